// EdgeTypeGAT_16260746183169
// MI455X (gfx1250) — compile-verified
//
#include <hip/hip_runtime.h>
#include <hip/hip_bf16.h>

typedef float v2f __attribute__((ext_vector_type(2)));
typedef float v8f __attribute__((ext_vector_type(8)));

#define HEADS 8
#define OUTF 16
#define CHAN 128   // HEADS*OUTF
#define INF_ 256

// ---------------------------------------------------------------------------
// Kernel 1: init out with bias, m with -inf, s with 0
// ---------------------------------------------------------------------------
__global__ __launch_bounds__(256) void k_init(float* __restrict__ out,
                                              float* __restrict__ m,
                                              float* __restrict__ s,
                                              const float* __restrict__ bias,
                                              int N) {
  int i = blockIdx.x * 256 + threadIdx.x;        // over N*128
  if (i >= N * CHAN) return;
  int c = i & (CHAN - 1);
  out[i] = bias[c];
  if (c < HEADS) {
    int j = (i >> 7) * HEADS + c;                // n*8 + h
    m[j] = __int_as_float(0xFF800000);           // -inf
    s[j] = 0.0f;
  }
}

// ---------------------------------------------------------------------------
// Kernel 2: feat = node_feat @ W via V_WMMA_F32_16X16X4_F32.
// One wave per 16-row block, computes all 8 column tiles (full 128 features),
// A fragment reused across the 8 B tiles. W staged through LDS in 2x64KB.
// ---------------------------------------------------------------------------
__global__ __launch_bounds__(256) void k_gemm(const float* __restrict__ A,
                                              const float* __restrict__ Wg,
                                              float* __restrict__ feat,
                                              int N, int nRowBlocks) {
  __shared__ float sW[128 * CHAN];               // 64 KB: 128 k-rows of W
  const int tid  = threadIdx.x;
  const int wave = tid >> 5;
  const int lane = tid & 31;
  const int half = lane >> 4;
  const int lrow = lane & 15;

  int rb = blockIdx.x * 8 + wave;
  const bool active = (rb < nRowBlocks);         // wave-uniform
  int rbc = active ? rb : (nRowBlocks - 1);
  int row = rbc * 16 + lrow;
  if (row >= N) row = N - 1;
  const float* Arow = A + (size_t)row * INF_;

  v8f acc[8] = {};

  for (int phase = 0; phase < 2; ++phase) {
    // cooperative, coalesced stage of 128 rows of W into LDS
    for (int i = tid; i < 128 * CHAN; i += 256)
      sW[i] = Wg[phase * 128 * CHAN + i];
    __syncthreads();

    const float* Ak = Arow + phase * 128;
    for (int kb = 0; kb < 32; ++kb) {
      const int k0 = kb * 4 + 2 * half;          // local k (0..127)
      v2f a;
      a.x = Ak[k0];
      a.y = Ak[k0 + 1];
#pragma unroll
      for (int t = 0; t < 8; ++t) {
        v2f b;
        b.x = sW[k0 * CHAN + t * 16 + lrow];
        b.y = sW[(k0 + 1) * CHAN + t * 16 + lrow];
        acc[t] = __builtin_amdgcn_wmma_f32_16x16x4_f32(
            false, a, false, b, (short)0, acc[t], false, false);
      }
    }
    __syncthreads();
  }

  if (active) {
    float* out = feat + (size_t)rb * 16 * CHAN;
#pragma unroll
    for (int t = 0; t < 8; ++t)
#pragma unroll
      for (int i = 0; i < 8; ++i)
        out[(size_t)(i + 8 * half) * CHAN + t * 16 + lrow] = acc[t][i];
  }
}

// ---------------------------------------------------------------------------
// Kernel 3: el/er per (node, head): dot-16 against attn_l / attn_r
// feat offset for pair i = n*8+h is exactly i*16.
// ---------------------------------------------------------------------------
__global__ __launch_bounds__(256) void k_attn(const float* __restrict__ feat,
                                              const float* __restrict__ attn_l,
                                              const float* __restrict__ attn_r,
                                              float* __restrict__ el,
                                              float* __restrict__ er,
                                              int NH) {
  int i = blockIdx.x * 256 + threadIdx.x;
  if (i >= NH) return;
  int h = i & (HEADS - 1);
  const float* f = feat + (size_t)i * OUTF;
  float sl = 0.f, sr = 0.f;
#pragma unroll
  for (int d = 0; d < OUTF; ++d) {
    float v = f[d];
    sl += v * attn_l[h * OUTF + d];
    sr += v * attn_r[h * OUTF + d];
  }
  el[i] = sl;
  er[i] = sr;
}

// ---------------------------------------------------------------------------
// Sign-aware float atomic-max via integer atomics (total order matches float
// for our -inf-initialized buffer).
// ---------------------------------------------------------------------------
__device__ __forceinline__ void atomicMaxF(float* addr, float val) {
  if (val >= 0.0f)
    atomicMax((int*)addr, __float_as_int(val));
  else
    atomicMin((unsigned int*)addr, (unsigned int)__float_as_int(val));
}

// ---------------------------------------------------------------------------
// Kernel 4: per-edge score e = leaky_relu(el[src]+er[dst]); segment max by dst
// ---------------------------------------------------------------------------
__global__ __launch_bounds__(256) void k_score(const float* __restrict__ el,
                                               const float* __restrict__ er,
                                               const int* __restrict__ src,
                                               const int* __restrict__ dst,
                                               float* __restrict__ ebuf,
                                               float* __restrict__ m,
                                               int E) {
  int e = blockIdx.x * 256 + threadIdx.x;
  if (e >= E) return;
  const int sn = src[e];
  const int dn = dst[e];
  const float* pl = el + (size_t)sn * HEADS;
  const float* pr = er + (size_t)dn * HEADS;
  float* pe = ebuf + (size_t)e * HEADS;
#pragma unroll
  for (int h = 0; h < HEADS; ++h) {
    float v = pl[h] + pr[h];
    v = (v > 0.f) ? v : 0.2f * v;                // LeakyReLU(0.2)
    pe[h] = v;
    atomicMaxF(&m[(size_t)dn * HEADS + h], v);
  }
}

// ---------------------------------------------------------------------------
// Kernel 5: a = exp(e - m[dst]); segment sum by dst
// ---------------------------------------------------------------------------
__global__ __launch_bounds__(256) void k_expsum(const int* __restrict__ dst,
                                                const float* __restrict__ m,
                                                float* __restrict__ ebuf,
                                                float* __restrict__ s,
                                                int E) {
  int e = blockIdx.x * 256 + threadIdx.x;
  if (e >= E) return;
  const int dn = dst[e];
  const float* pm = m + (size_t)dn * HEADS;
  float* pe = ebuf + (size_t)e * HEADS;
#pragma unroll
  for (int h = 0; h < HEADS; ++h) {
    float a = __expf(pe[h] - pm[h]);
    pe[h] = a;
    atomicAdd(&s[(size_t)dn * HEADS + h], a);
  }
}

// ---------------------------------------------------------------------------
// Kernel 6: weighted scatter-aggregate. 128 lanes per edge (one per channel):
// coalesced 512B gather of feat[src], coalesced 512B atomicAdd into out[dst].
// ---------------------------------------------------------------------------
__global__ __launch_bounds__(256) void k_aggregate(const float* __restrict__ feat,
                                                   const float* __restrict__ ebuf,
                                                   const float* __restrict__ s,
                                                   const int* __restrict__ src,
                                                   const int* __restrict__ dst,
                                                   float* __restrict__ out,
                                                   int E) {
  int e = blockIdx.x * 2 + (threadIdx.x >> 7);
  if (e >= E) return;
  int c = threadIdx.x & (CHAN - 1);
  int h = c >> 4;
  const int sn = src[e];
  const int dn = dst[e];
  float alpha = ebuf[(size_t)e * HEADS + h] / s[(size_t)dn * HEADS + h];
  atomicAdd(&out[(size_t)dn * CHAN + c],
            alpha * feat[(size_t)sn * CHAN + c]);
}

// ---------------------------------------------------------------------------
extern "C" void kernel_launch(void* const* d_in, const int* in_sizes, int n_in,
                              void* d_out, int out_size, void* d_ws, size_t ws_size,
                              hipStream_t stream) {
  const float* node_feat = (const float*)d_in[0];
  const int*   src       = (const int*)d_in[1];
  const int*   dst       = (const int*)d_in[2];
  const float* W         = (const float*)d_in[3];
  const float* attn_l    = (const float*)d_in[4];
  const float* attn_r    = (const float*)d_in[5];
  const float* bias      = (const float*)d_in[6];

  const int N = in_sizes[0] / INF_;   // 100000
  const int E = in_sizes[1];          // 1600000
  float* out = (float*)d_out;

  // workspace layout
  float* feat = (float*)d_ws;                      // N*128
  float* el   = feat + (size_t)N * CHAN;           // N*8
  float* er   = el   + (size_t)N * HEADS;          // N*8
  float* m    = er   + (size_t)N * HEADS;          // N*8
  float* s    = m    + (size_t)N * HEADS;          // N*8
  float* ebuf = s    + (size_t)N * HEADS;          // E*8

  const int nRowBlocks = (N + 15) / 16;

  // 1) init out=bias, m=-inf, s=0
  k_init<<<(N * CHAN + 255) / 256, 256, 0, stream>>>(out, m, s, bias, N);
  // 2) WMMA GEMM: feat = node_feat @ W
  k_gemm<<<(nRowBlocks + 7) / 8, 256, 0, stream>>>(node_feat, W, feat, N, nRowBlocks);
  // 3) el/er
  k_attn<<<(N * HEADS + 255) / 256, 256, 0, stream>>>(feat, attn_l, attn_r, el, er, N * HEADS);
  // 4) edge scores + segment max
  k_score<<<(E + 255) / 256, 256, 0, stream>>>(el, er, src, dst, ebuf, m, E);
  // 5) exp + segment sum
  k_expsum<<<(E + 255) / 256, 256, 0, stream>>>(dst, m, ebuf, s, E);
  // 6) normalize + scatter aggregate
  k_aggregate<<<(E + 1) / 2, 256, 0, stream>>>(feat, ebuf, s, src, dst, out, E);
}